// GNN_54949811585355
// MI455X (gfx1250) — compile-verified
//
#include <hip/hip_runtime.h>

#define HID    128
#define INDIM  130
#define LDA    132          // LDS row pitch (floats): bank-conflict-free for b64 frag loads
#define LN_EPS 1e-5f

typedef __attribute__((ext_vector_type(2))) float v2f;
typedef __attribute__((ext_vector_type(8))) float v8f;

// ---------------------------------------------------------------------------
// Zero a float4-aligned region (grid-stride).
// ---------------------------------------------------------------------------
__global__ void gnn_zero_kernel(float4* __restrict__ p, long n4) {
  long i      = (long)blockIdx.x * blockDim.x + threadIdx.x;
  long stride = (long)gridDim.x * blockDim.x;
  float4 z = {0.f, 0.f, 0.f, 0.f};
  for (; i < n4; i += stride) p[i] = z;
}

// ---------------------------------------------------------------------------
// Edge scatter: one wave32 per edge; lanes stride the feature dim.
// agg[dst][:] += feat[src][:]  (atomic, L2-resident accumulator)
// deg[dst]    += 1             (layer-1 pass only)
// ---------------------------------------------------------------------------
__global__ void __launch_bounds__(256) gnn_scatter_kernel(
    const float* __restrict__ feat,
    const int*   __restrict__ src,
    const int*   __restrict__ dst,
    float*       __restrict__ agg,
    float*       __restrict__ deg,
    int nedges, int dim) {
  int gwave = (int)(((long)blockIdx.x * blockDim.x + threadIdx.x) >> 5);
  int lane  = threadIdx.x & 31;
  if (gwave >= nedges) return;
  int s = src[gwave];
  int d = dst[gwave];
  const float* frow = feat + (long)s * dim;
  float*       arow = agg  + (long)d * dim;
  for (int f = lane; f < dim; f += 32)
    atomicAdd(&arow[f], frow[f]);
  if (deg != nullptr && lane == 0)
    atomicAdd(&deg[d], 1.0f);
}

// ---------------------------------------------------------------------------
// Cooperative stage of a 16 x KDIM tile (row-major, pitch LDA) into LDS.
// ---------------------------------------------------------------------------
template <int KDIM>
__device__ __forceinline__ void gnn_stage_tile(
    float* __restrict__ s, const float* __restrict__ g, int m0, int tid) {
  for (int e = tid; e < 16 * KDIM; e += 256) {
    int r = e / KDIM;
    int c = e - r * KDIM;
    s[r * LDA + c] = g[(long)(m0 + r) * KDIM + c];
  }
}

// ---------------------------------------------------------------------------
// C += (ascale * A_row(16xK)) * W^T(Kx16) via V_WMMA_F32_16X16X4_F32.
// KDIM compile-time: main loop branch-free; 2-wide tail (KDIM=130) is one
// extra WMMA where only the koff==0 lane-half carries data.
//   A frag: lane l -> A[lane&15][k + (l>>4)*2 + {0,1}]
//   B frag: lane l -> W[n0+(lane&15)][k + (l>>4)*2 + {0,1}]
// arow (LDS) / wrow (global) are pre-offset to this lane's row.
// ---------------------------------------------------------------------------
template <int KDIM, int AROW_PITCH_UNUSED = 0>
__device__ __forceinline__ v8f gnn_wmma_mac(
    v8f c,
    const float* __restrict__ arow, float ascale,
    const float* __restrict__ wrow, int koff) {
  const int KMAIN = KDIM & ~3;
#pragma unroll
  for (int k = 0; k < KMAIN; k += 4) {
    v2f a = *(const v2f*)(arow + k + koff);
    v2f b = *(const v2f*)(wrow + k + koff);
    a *= ascale;
    c = __builtin_amdgcn_wmma_f32_16x16x4_f32(
        false, a, false, b, (short)0, c, false, false);
  }
  if (KDIM != KMAIN) {
    v2f a = {0.f, 0.f};
    v2f b = {0.f, 0.f};
    if (koff == 0) {
      a = *(const v2f*)(arow + KMAIN);
      b = *(const v2f*)(wrow + KMAIN);
    }
    a *= ascale;
    c = __builtin_amdgcn_wmma_f32_16x16x4_f32(
        false, a, false, b, (short)0, c, false, false);
  }
  return c;
}

// ---------------------------------------------------------------------------
// Layer 1: out = (agg/deg) @ W1l^T + b1 + x @ W1r^T, then LayerNorm + ReLU.
// Block = 256 thr = 8 waves; block owns 16 nodes; wave w owns cols [16w,16w+16).
// A tiles staged in LDS once per block; weights stream from global (L2-hot).
// ---------------------------------------------------------------------------
__global__ void __launch_bounds__(256) gnn_gemm1_ln_relu_kernel(
    const float* __restrict__ x,   const float* __restrict__ agg,
    const float* __restrict__ deg,
    const float* __restrict__ W1l, const float* __restrict__ b1l,
    const float* __restrict__ W1r,
    const float* __restrict__ lng, const float* __restrict__ lnb,
    float* __restrict__ h) {
  __shared__ float sA[16 * LDA];              // agg tile
  __shared__ float sX[16 * LDA];              // x tile
  __shared__ float sO[16 * LDA];              // output tile for LN

  const int tid  = threadIdx.x;
  const int lane = tid & 31;
  const int wave = tid >> 5;
  const int m0   = blockIdx.x * 16;
  const int n0   = wave * 16;
  const int mi   = lane & 15;
  const int koff = (lane >> 4) * 2;
  const int ncol = n0 + mi;

  gnn_stage_tile<INDIM>(sA, agg, m0, tid);
  gnn_stage_tile<INDIM>(sX, x,   m0, tid);
  const float rdeg = 1.0f / fmaxf(deg[m0 + mi], 1.0f);
  __syncthreads();

  v8f c = {};
  c = gnn_wmma_mac<INDIM>(c, sA + mi * LDA, rdeg,
                          W1l + (long)ncol * INDIM, koff);
  c = gnn_wmma_mac<INDIM>(c, sX + mi * LDA, 1.0f,
                          W1r + (long)ncol * INDIM, koff);

  // C layout: VGPR r, lanes 0-15 -> M=r; lanes 16-31 -> M=r+8; N = ncol.
  const float bias  = b1l[ncol];
  const int   rbase = (lane >> 4) * 8;
#pragma unroll
  for (int r = 0; r < 8; ++r)
    sO[(rbase + r) * LDA + ncol] = c[r] + bias;
  __syncthreads();

  // LayerNorm + ReLU: 16 threads per row, 8 values each, shfl-xor butterfly.
  const int row = tid >> 4;                   // 0..15
  const int j   = tid & 15;                   // 0..15
  const float* srow = sO + row * LDA + j * 8;
  float v[8];
  float s = 0.f, sq = 0.f;
#pragma unroll
  for (int i = 0; i < 8; ++i) { v[i] = srow[i]; s += v[i]; sq += v[i] * v[i]; }
#pragma unroll
  for (int m = 1; m < 16; m <<= 1) {
    s  += __shfl_xor(s,  m, 16);
    sq += __shfl_xor(sq, m, 16);
  }
  const float mu   = s * (1.0f / HID);
  const float var  = sq * (1.0f / HID) - mu * mu;
  const float rstd = rsqrtf(var + LN_EPS);

  float* orow = h + (long)(m0 + row) * HID + j * 8;
#pragma unroll
  for (int i = 0; i < 8; ++i) {
    int col = j * 8 + i;
    float y = (v[i] - mu) * rstd * lng[col] + lnb[col];
    orow[i] = fmaxf(y, 0.0f);
  }
}

// ---------------------------------------------------------------------------
// Layer 2: out = (agg/deg) @ W2l^T + b2 + h @ W2r^T  -> d_out
// ---------------------------------------------------------------------------
__global__ void __launch_bounds__(256) gnn_gemm2_kernel(
    const float* __restrict__ h,   const float* __restrict__ agg,
    const float* __restrict__ deg,
    const float* __restrict__ W2l, const float* __restrict__ b2l,
    const float* __restrict__ W2r,
    float* __restrict__ out) {
  __shared__ float sA[16 * LDA];
  __shared__ float sX[16 * LDA];

  const int tid  = threadIdx.x;
  const int lane = tid & 31;
  const int wave = tid >> 5;
  const int m0   = blockIdx.x * 16;
  const int n0   = wave * 16;
  const int mi   = lane & 15;
  const int koff = (lane >> 4) * 2;
  const int ncol = n0 + mi;

  gnn_stage_tile<HID>(sA, agg, m0, tid);
  gnn_stage_tile<HID>(sX, h,   m0, tid);
  const float rdeg = 1.0f / fmaxf(deg[m0 + mi], 1.0f);
  __syncthreads();

  v8f c = {};
  c = gnn_wmma_mac<HID>(c, sA + mi * LDA, rdeg,
                        W2l + (long)ncol * HID, koff);
  c = gnn_wmma_mac<HID>(c, sX + mi * LDA, 1.0f,
                        W2r + (long)ncol * HID, koff);

  const float bias  = b2l[ncol];
  const int   rbase = (lane >> 4) * 8;
#pragma unroll
  for (int r = 0; r < 8; ++r)
    out[(long)(m0 + rbase + r) * HID + ncol] = c[r] + bias;
}

// ---------------------------------------------------------------------------
// Launch: zero -> scatter1(+deg) -> gemm1+LN+ReLU -> zero -> scatter2 -> gemm2
// Workspace layout: [agg: N*INDIM][deg: N][h: N*HID]  (agg reused for layer 2)
// ---------------------------------------------------------------------------
extern "C" void kernel_launch(void* const* d_in, const int* in_sizes, int n_in,
                              void* d_out, int out_size, void* d_ws, size_t ws_size,
                              hipStream_t stream) {
  const float* x   = (const float*)d_in[0];
  const int*   ei  = (const int*)  d_in[1];
  const float* W1l = (const float*)d_in[2];
  const float* b1l = (const float*)d_in[3];
  const float* W1r = (const float*)d_in[4];
  const float* lng = (const float*)d_in[5];
  const float* lnb = (const float*)d_in[6];
  const float* W2l = (const float*)d_in[7];
  const float* b2l = (const float*)d_in[8];
  const float* W2r = (const float*)d_in[9];

  const int N = in_sizes[0] / INDIM;   // 100000
  const int E = in_sizes[1] / 2;       // 1600000
  const int* src = ei;
  const int* dst = ei + E;

  float* agg = (float*)d_ws;
  float* deg = agg + (size_t)N * INDIM;
  float* h   = deg + N;
  float* out = (float*)d_out;

  dim3 blk(256);
  const int sblocks = (E + 7) / 8;     // one wave32 per edge
  const int gblocks = N / 16;          // exact: 6250

  // Layer 1
  long z1 = ((long)N * INDIM + (long)N) / 4;      // agg + deg contiguous
  gnn_zero_kernel<<<4096, blk, 0, stream>>>((float4*)agg, z1);
  gnn_scatter_kernel<<<sblocks, blk, 0, stream>>>(x, src, dst, agg, deg, E, INDIM);
  gnn_gemm1_ln_relu_kernel<<<gblocks, blk, 0, stream>>>(
      x, agg, deg, W1l, b1l, W1r, lng, lnb, h);

  // Layer 2 (reuse agg buffer, deg unchanged)
  long z2 = ((long)N * HID) / 4;
  gnn_zero_kernel<<<4096, blk, 0, stream>>>((float4*)agg, z2);
  gnn_scatter_kernel<<<sblocks, blk, 0, stream>>>(h, src, dst, agg, nullptr, E, HID);
  gnn_gemm2_kernel<<<gblocks, blk, 0, stream>>>(h, agg, deg, W2l, b2l, W2r, out);
}